// LinearOperatorInterp_16982300688598
// MI455X (gfx1250) — compile-verified
//
#include <hip/hip_runtime.h>
#include <hip/hip_bf16.h>

// ---------------------------------------------------------------------------
// GP linear-operator interpolation layer, CDNA5 (gfx1250) implementation.
// bs=4, n=m=576, cin=32, cout=64, k=4 (id, d/dx, d/dy, lap), D=2.
// Heavy GEMMs: V_WMMA_F32_16X16X4_F32 with GLOBAL_LOAD_ASYNC_TO_LDS_B128
// double-buffered tile staging (ASYNCcnt-tracked, overlaps DMA with WMMA).
// ---------------------------------------------------------------------------

#define LSCALE 0.5414f
#define L2C (LSCALE * LSCALE)          // 0.29311396
#define JIT_KXX 1e-5f
#define JIT_SUM (1e-4f + 1e-5f)        // JITTER_AKA + JITTER_YY on final AKA

#ifndef USE_ASYNC
#define USE_ASYNC 1                    // flip to 0 if async mnemonic rejected
#endif

typedef float v2f __attribute__((ext_vector_type(2)));
typedef float v8f __attribute__((ext_vector_type(8)));

// Low 32 bits of a generic LDS pointer = wave-relative LDS byte address.
__device__ __forceinline__ unsigned lds_addr(const void* p) {
  return (unsigned)(unsigned long long)p;
}

#if USE_ASYNC
__device__ __forceinline__ void g2l_b128(const float* g, const float* l) {
  asm volatile("global_load_async_to_lds_b128 %0, %1, off"
               :: "v"(lds_addr(l)), "v"(g) : "memory");
}
template <int N>
__device__ __forceinline__ void wait_async() {
#if __has_builtin(__builtin_amdgcn_s_wait_asynccnt)
  __builtin_amdgcn_s_wait_asynccnt(N);
#else
  asm volatile("s_wait_asynccnt %0" :: "i"(N) : "memory");
#endif
}
#else
__device__ __forceinline__ void g2l_b128(const float* g, const float* l) {
  float4 v = *(const float4*)g;
  *(float4*)l = v;
}
template <int N>
__device__ __forceinline__ void wait_async() {}
#endif

// ---------------------------------------------------------------------------
// Kxx = exp(-0.5*|xi-xj|^2/l2) + jitter*I
// ---------------------------------------------------------------------------
__global__ void k_kxx(const float* __restrict__ x, float* __restrict__ Kxx, int n) {
  int b = blockIdx.z;
  int j = blockIdx.x * 16 + threadIdx.x;
  int i = blockIdx.y * 16 + threadIdx.y;
  if (i >= n || j >= n) return;
  const float* xb = x + (size_t)b * n * 2;
  float dx = xb[i * 2 + 0] - xb[j * 2 + 0];
  float dy = xb[i * 2 + 1] - xb[j * 2 + 1];
  float u = (dx * dx + dy * dy) / L2C;
  float v = expf(-0.5f * u);
  if (i == j) v += JIT_KXX;
  Kxx[((size_t)b * n + i) * n + j] = v;
}

// ---------------------------------------------------------------------------
// opiK planes (b, k, m, n): [K, -r0/l2*K, -r1/l2*K, ((u-d)/l2)*K]
// ---------------------------------------------------------------------------
__global__ void k_opik(const float* __restrict__ x, float* __restrict__ out, int n) {
  int z = blockIdx.z; int b = z >> 2, i0 = z & 3;
  int q  = blockIdx.x * 16 + threadIdx.x;
  int mm = blockIdx.y * 16 + threadIdx.y;
  if (q >= n || mm >= n) return;
  const float* xb = x + (size_t)b * n * 2;
  float r0 = xb[mm * 2 + 0] - xb[q * 2 + 0];
  float r1 = xb[mm * 2 + 1] - xb[q * 2 + 1];
  float il2 = 1.0f / L2C;
  float u = (r0 * r0 + r1 * r1) * il2;
  float K = expf(-0.5f * u);
  float v;
  if (i0 == 0)      v = K;
  else if (i0 == 1) v = -r0 * il2 * K;
  else if (i0 == 2) v = -r1 * il2 * K;
  else              v = (u - 2.0f) * il2 * K;
  out[((size_t)z * n + mm) * n + q] = v;
}

// ---------------------------------------------------------------------------
// In-place batched Cholesky (lower), one workgroup per batch, LDS column cache
// ---------------------------------------------------------------------------
__global__ __launch_bounds__(1024) void k_cholesky(float* __restrict__ A, int n) {
  __shared__ float col[640];
  int b = blockIdx.x;
  A += (size_t)b * n * n;
  int tid = threadIdx.x, nt = blockDim.x;
  for (int j = 0; j < n; ++j) {
    if (tid == 0) {
      float d = A[(size_t)j * n + j];
      d = sqrtf(fmaxf(d, 1e-20f));
      A[(size_t)j * n + j] = d;
      col[j] = d;
    }
    __syncthreads();
    float dinv = 1.0f / col[j];
    for (int i = j + 1 + tid; i < n; i += nt) {
      float v = A[(size_t)i * n + j] * dinv;
      A[(size_t)i * n + j] = v;
      col[i] = v;
    }
    __syncthreads();
    int rem = n - j - 1;
    int cnt = rem * rem;
    for (int idx = tid; idx < cnt; idx += nt) {
      int i = j + 1 + idx / rem;
      int q = j + 1 + idx % rem;
      if (q <= i) A[(size_t)i * n + q] -= col[i] * col[q];
    }
    __syncthreads();
  }
}

// ---------------------------------------------------------------------------
// Triangular solves, one RHS column per thread (in-place on B, n x r row-major)
// L row loads are wave-uniform (scalar cache), B accesses coalesced.
// ---------------------------------------------------------------------------
__global__ void k_trsm_lo(const float* __restrict__ L, float* __restrict__ B,
                          int n, int r, size_t sL, size_t sB) {
  int b = blockIdx.z;
  L += (size_t)b * sL; B += (size_t)b * sB;
  int c = blockIdx.x * blockDim.x + threadIdx.x;
  if (c >= r) return;
  for (int i = 0; i < n; ++i) {
    const float* Li = L + (size_t)i * n;
    float s = B[(size_t)i * r + c];
    for (int q = 0; q < i; ++q) s -= Li[q] * B[(size_t)q * r + c];
    B[(size_t)i * r + c] = s / Li[i];
  }
}

__global__ void k_trsm_up(const float* __restrict__ L, float* __restrict__ B,
                          int n, int r, size_t sL, size_t sB) {
  int b = blockIdx.z;
  L += (size_t)b * sL; B += (size_t)b * sB;
  int c = blockIdx.x * blockDim.x + threadIdx.x;
  if (c >= r) return;
  for (int i = n - 1; i >= 0; --i) {
    float s = B[(size_t)i * r + c];
    for (int q = i + 1; q < n; ++q)
      s -= L[(size_t)q * n + i] * B[(size_t)q * r + c];
    B[(size_t)i * r + c] = s / L[(size_t)i * n + i];
  }
}

// ---------------------------------------------------------------------------
// Xbuf[b][q][i0*m+mm] = opiK[b][i0][mm][q]  (tiled transpose per plane)
// ---------------------------------------------------------------------------
__global__ void k_pack_xT(const float* __restrict__ opiK, float* __restrict__ X,
                          int m, int n, int k) {
  __shared__ float tile[32][33];
  int z = blockIdx.z; int b = z / k, i0 = z % k;
  const float* src = opiK + (size_t)z * m * n;
  float* dst = X + (size_t)b * n * (size_t)(k * m);
  int x0 = blockIdx.x * 32, y0 = blockIdx.y * 32;
  int tx = threadIdx.x, ty = threadIdx.y;
  for (int yy = ty; yy < 32; yy += 8) {
    int mm = y0 + yy, q = x0 + tx;
    tile[yy][tx] = (mm < m && q < n) ? src[(size_t)mm * n + q] : 0.0f;
  }
  __syncthreads();
  for (int yy = ty; yy < 32; yy += 8) {
    int q = x0 + yy, mm = y0 + tx;
    if (q < n && mm < m)
      dst[(size_t)q * (k * m) + (size_t)i0 * m + mm] = tile[tx][yy];
  }
}

// ---------------------------------------------------------------------------
// T = S^T - I   (S = Kxx^-1 Kin)
// ---------------------------------------------------------------------------
__global__ void k_T_from_S(const float* __restrict__ S, float* __restrict__ T, int n) {
  __shared__ float tile[32][33];
  int b = blockIdx.z;
  S += (size_t)b * n * n; T += (size_t)b * n * n;
  int x0 = blockIdx.x * 32, y0 = blockIdx.y * 32;
  int tx = threadIdx.x, ty = threadIdx.y;
  for (int yy = ty; yy < 32; yy += 8) {
    int rr = y0 + yy, cc = x0 + tx;
    tile[yy][tx] = (rr < n && cc < n) ? S[(size_t)rr * n + cc] : 0.0f;
  }
  __syncthreads();
  for (int yy = ty; yy < 32; yy += 8) {
    int q = x0 + yy, nn = y0 + tx;
    if (q < n && nn < n) {
      float v = tile[tx][yy];
      if (q == nn) v -= 1.0f;
      T[(size_t)q * n + nn] = v;
    }
  }
}

// ---------------------------------------------------------------------------
// W2[i][j] = (1/cout) * sum_{o,c} W[o][c*k+i] * W[o][c*k+j]
// ---------------------------------------------------------------------------
__global__ void k_w2(const float* __restrict__ W, float* __restrict__ W2,
                     int cout, int cin, int k) {
  int t = threadIdx.x;
  if (t >= k * k) return;
  int i = t / k, j = t % k;
  float s = 0.0f;
  for (int o = 0; o < cout; ++o)
    for (int c = 0; c < cin; ++c)
      s += W[o * (cin * k) + c * k + i] * W[o * (cin * k) + c * k + j];
  W2[t] = s / (float)cout;
}

// ---------------------------------------------------------------------------
// U[b][j][mm][q] = sum_i W2[i][j] * X[b][q][i*m+mm]   (tiled transpose)
// ---------------------------------------------------------------------------
__global__ void k_build_U(const float* __restrict__ X, const float* __restrict__ W2,
                          float* __restrict__ U, int m, int n, int k) {
  __shared__ float tile[16][17];
  int z = blockIdx.z; int b = z / k, j = z % k;
  const float* Xb = X + (size_t)b * n * (size_t)(k * m);
  float* Uo = U + (size_t)z * m * n;
  float w0 = W2[0 * k + j], w1 = W2[1 * k + j], w2 = W2[2 * k + j], w3 = W2[3 * k + j];
  int q0 = blockIdx.x * 16, mm0 = blockIdx.y * 16;
  int tx = threadIdx.x, ty = threadIdx.y;
  int q = q0 + ty, mm = mm0 + tx;
  float acc = 0.0f;
  if (q < n && mm < m) {
    const float* row = Xb + (size_t)q * (k * m);
    acc = w0 * row[mm] + w1 * row[m + mm] + w2 * row[2 * m + mm] + w3 * row[3 * m + mm];
  }
  tile[ty][tx] = acc;
  __syncthreads();
  int mmw = mm0 + ty, qw = q0 + tx;
  if (mmw < m && qw < n) Uo[(size_t)mmw * n + qw] = tile[tx][ty];
}

// ---------------------------------------------------------------------------
// Closed-form  sum_ij W2[ij]*opiKopjfull[ij](x_mm, x_p)  + jitter*I  -> AKA base
// ---------------------------------------------------------------------------
__global__ void k_copj(const float* __restrict__ x, const float* __restrict__ W2,
                       float* __restrict__ AKA, int n) {
  int b = blockIdx.z;
  int p  = blockIdx.x * 16 + threadIdx.x;
  int mm = blockIdx.y * 16 + threadIdx.y;
  if (p >= n || mm >= n) return;
  const float* xb = x + (size_t)b * n * 2;
  float r0 = xb[mm * 2 + 0] - xb[p * 2 + 0];
  float r1 = xb[mm * 2 + 1] - xb[p * 2 + 1];
  float il2 = 1.0f / L2C, il4 = il2 * il2;
  float u = (r0 * r0 + r1 * r1) * il2;
  float K = expf(-0.5f * u);
  const float d = 2.0f;
  float g0 = r0 * il2 * K, g1 = r1 * il2 * K;       // d/dy_j K
  float lap = (u - d) * il2 * K;
  float gg00 = (il2 - r0 * r0 * il4) * K;
  float gg01 = (-r0 * r1 * il4) * K;
  float gg11 = (il2 - r1 * r1 * il4) * K;
  float gl0 = r0 * il4 * (2.0f + d - u) * K;        // d/dx_i Lap_y K
  float gl1 = r1 * il4 * (2.0f + d - u) * K;
  float ll = ((u - d) * (u - d) - 4.0f * u + 2.0f * d) * il4 * K;
  float s = W2[0] * K
          + W2[1] * g0 + W2[2] * g1 + W2[3] * lap
          + W2[4] * (-g0) + W2[8] * (-g1) + W2[12] * lap
          + W2[5] * gg00 + W2[6] * gg01 + W2[9] * gg01 + W2[10] * gg11
          + W2[7] * gl0 + W2[11] * gl1
          + W2[13] * (-gl0) + W2[14] * (-gl1)
          + W2[15] * ll;
  if (mm == p) s += JIT_SUM;
  AKA[((size_t)b * n + mm) * n + p] = s;
}

// ---------------------------------------------------------------------------
// Amean[b][mm][o] = bias[o] + sum_{c,i} opimean[b][i][mm][c] * W[o][c*k+i]
// ---------------------------------------------------------------------------
__global__ void k_amean(const float* __restrict__ om, const float* __restrict__ W,
                        const float* __restrict__ bias, float* __restrict__ out,
                        int bs, int m, int c, int k, int cout) {
  int id = blockIdx.x * blockDim.x + threadIdx.x;
  int total = bs * m * cout;
  if (id >= total) return;
  int o = id % cout;
  int mm = (id / cout) % m;
  int b = id / (cout * m);
  float s = bias[o];
  for (int cc = 0; cc < c; ++cc)
    for (int i = 0; i < k; ++i)
      s += om[(((size_t)(b * k + i)) * m + mm) * c + cc] * W[o * (c * k) + cc * k + i];
  out[((size_t)b * m + mm) * cout + o] = s;
}

// ---------------------------------------------------------------------------
// Batched fp32 WMMA GEMM, branch-free tiles: C = (ACC ? C : 0) + A @ opB(B)
// Requirements (met by all call sites): M % 64 == 0, K % 16 == 0.
// N may be ragged: extra B columns read benign in-workspace data consumed
// only by lanes whose stores are masked (gcol < N).
//   A: (M x K) row-major.  !TB: B (K x N) row-major.  TB: B (N x K) row-major.
// 64x64 macro tile, 256 threads = 8 waves, wave = 32x16 (two 16x16 accs).
// Tiles staged via async b128 -> LDS, double buffered (ASYNCcnt-tracked).
// ---------------------------------------------------------------------------
template <bool TB, bool ACC>
__global__ __launch_bounds__(256)
void k_gemm(const float* __restrict__ Ag, const float* __restrict__ Bg,
            float* __restrict__ Cg, int M, int N, int K,
            int lda, int ldb, int ldc, long sA, long sB, long sC) {
  constexpr int ASTR = 20;              // 64 rows x 16 K, pad 20: conflict-free
  constexpr int BSTR = TB ? 20 : 68;    // TB: n-major 64x16; !TB: k-major 16x64
  __shared__ float As[2][64 * ASTR];
  __shared__ float Bs[2][TB ? 64 * 20 : 16 * 68];
  int z = blockIdx.z;
  Ag += (size_t)z * sA; Bg += (size_t)z * sB; Cg += (size_t)z * sC;
  int m0 = blockIdx.y * 64;
  int n0 = blockIdx.x * 64;
  int tid = threadIdx.x;
  int lane = tid & 31, wave = tid >> 5;
  int wm = wave & 1, wn = wave >> 1;    // 2x4 wave grid over 64x64
  int half = lane >> 4, rl = lane & 15;

  // global/LDS staging map: one 16-byte chunk per thread per operand per tile
  int aRow = tid >> 2, aC4 = (tid & 3) * 4;
  const float* agp = Ag + (size_t)(m0 + aRow) * lda + aC4;    // advance +16/step
  const float* alp0 = &As[0][aRow * ASTR + aC4];
  const float* alp1 = &As[1][aRow * ASTR + aC4];
  int bRow, bC4;
  long bstep;
  if (!TB) { bRow = tid >> 4; bC4 = (tid & 15) * 4; bstep = (long)16 * ldb; }
  else     { bRow = tid >> 2; bC4 = (tid & 3) * 4;  bstep = 16; }
  const float* bgp = TB ? Bg + (size_t)(n0 + bRow) * ldb + bC4
                        : Bg + (size_t)bRow * ldb + n0 + bC4;
  const float* blp0 = &Bs[0][bRow * BSTR + bC4];
  const float* blp1 = &Bs[1][bRow * BSTR + bC4];

  v8f c0 = {}, c1 = {};
  int nk = K >> 4;
  // prologue: tile 0 -> buffer 0
  g2l_b128(agp, alp0);
  g2l_b128(bgp, blp0);
  for (int t = 0; t < nk; ++t) {
    int cur = t & 1;
    if (t + 1 < nk) {                   // prefetch tile t+1 into other buffer
      const float* ga = agp + (size_t)(t + 1) * 16;
      const float* gb = bgp + (size_t)(t + 1) * bstep;
      g2l_b128(ga, cur ? alp0 : alp1);
      g2l_b128(gb, cur ? blp0 : blp1);
      wait_async<2>();                  // tile t complete; t+1 in flight
    } else {
      wait_async<0>();
    }
    __syncthreads();                    // tile t visible to all waves
    const float* Ab = As[cur];
    const float* Bb = Bs[cur];
#pragma unroll
    for (int kk = 0; kk < 4; ++kk) {
      // ISA 7.12.2: A 16x4 frag: row = lane&15, VGPR pair K = 2*(lane>>4)..+1
      // B 4x16 frag: row-striped across lanes, same K pairing.
      int kb = kk * 4 + half * 2;
      int ar0 = wm * 32 + rl;
      v2f a0, a1, bf;
      a0[0] = Ab[ar0 * ASTR + kb];        a0[1] = Ab[ar0 * ASTR + kb + 1];
      a1[0] = Ab[(ar0 + 16) * ASTR + kb]; a1[1] = Ab[(ar0 + 16) * ASTR + kb + 1];
      int bc = wn * 16 + rl;
      if (!TB) { bf[0] = Bb[kb * BSTR + bc]; bf[1] = Bb[(kb + 1) * BSTR + bc]; }
      else     { bf[0] = Bb[bc * BSTR + kb]; bf[1] = Bb[bc * BSTR + kb + 1]; }
      c0 = __builtin_amdgcn_wmma_f32_16x16x4_f32(false, a0, false, bf,
                                                 (short)0, c0, false, false);
      c1 = __builtin_amdgcn_wmma_f32_16x16x4_f32(false, a1, false, bf,
                                                 (short)0, c1, false, false);
    }
    __syncthreads();                    // all waves done with buf before reuse
  }
  // C/D layout: VGPR e, lanes 0-15 -> (M=e, N=lane), lanes 16-31 -> (M=e+8)
  int gcol = n0 + wn * 16 + rl;
  if (gcol < N) {
#pragma unroll
    for (int e = 0; e < 8; ++e) {
      int gr0 = m0 + wm * 32 + e + half * 8;
      size_t o0 = (size_t)gr0 * ldc + gcol;
      Cg[o0] = (ACC ? Cg[o0] : 0.0f) + c0[e];
      size_t o1 = (size_t)(gr0 + 16) * ldc + gcol;
      Cg[o1] = (ACC ? Cg[o1] : 0.0f) + c1[e];
    }
  }
}

// ---------------------------------------------------------------------------
extern "C" void kernel_launch(void* const* d_in, const int* in_sizes, int n_in,
                              void* d_out, int out_size, void* d_ws, size_t ws_size,
                              hipStream_t stream) {
  const float* xin    = (const float*)d_in[0];
  const float* meanin = (const float*)d_in[1];
  const float* Kin    = (const float*)d_in[2];
  const float* weight = (const float*)d_in[3];
  const float* bias   = (const float*)d_in[4];

  const int k = 4;
  int cout = in_sizes[4];
  int n    = 2 * (in_sizes[2] / in_sizes[0]);        // Kin/xin = n/2
  int bs   = in_sizes[0] / (n * 2);
  int cin  = (in_sizes[3] / cout) / k;
  const int m = n, km = k * m;

  float* ws = (float*)d_ws;
  size_t oKxx  = 0;
  size_t oOpiK = oKxx  + (size_t)bs * n * n;
  size_t oX    = oOpiK + (size_t)bs * km * n;        // X; later reused as R
  size_t oS    = oX    + (size_t)bs * n * km;
  size_t oT    = oS    + (size_t)bs * n * n;
  size_t oSM   = oT    + (size_t)bs * n * n;
  size_t oOM   = oSM   + (size_t)bs * n * cin;
  size_t oU    = oOM   + (size_t)bs * km * cin;
  size_t oW2   = oU    + (size_t)bs * km * n;
  float* Rbuf = ws + oX;                              // alias: X dead after U

  float* outx     = (float*)d_out;
  float* outAmean = outx + (size_t)bs * n * 2;
  float* outAKA   = outAmean + (size_t)bs * m * cout;

  dim3 b16(16, 16), b32x8(32, 8);

  // xout = xin
  hipMemcpyAsync(outx, xin, sizeof(float) * (size_t)bs * n * 2,
                 hipMemcpyDeviceToDevice, stream);
  // W2
  k_w2<<<1, 32, 0, stream>>>(weight, ws + oW2, cout, cin, k);
  // Kxx + jitter, Cholesky in place
  k_kxx<<<dim3((n + 15) / 16, (n + 15) / 16, bs), b16, 0, stream>>>(xin, ws + oKxx, n);
  k_cholesky<<<bs, 1024, 0, stream>>>(ws + oKxx, n);
  // opiK planes
  k_opik<<<dim3((n + 15) / 16, (m + 15) / 16, bs * k), b16, 0, stream>>>(xin, ws + oOpiK, n);
  // solved mean = Kxx^-1 meanin
  hipMemcpyAsync(ws + oSM, meanin, sizeof(float) * (size_t)bs * n * cin,
                 hipMemcpyDeviceToDevice, stream);
  {
    dim3 g((cin + 255) / 256, 1, bs);
    k_trsm_lo<<<g, 256, 0, stream>>>(ws + oKxx, ws + oSM, n, cin,
                                     (size_t)n * n, (size_t)n * cin);
    k_trsm_up<<<g, 256, 0, stream>>>(ws + oKxx, ws + oSM, n, cin,
                                     (size_t)n * n, (size_t)n * cin);
  }
  // S = Kxx^-1 Kin ; T = S^T - I
  hipMemcpyAsync(ws + oS, Kin, sizeof(float) * (size_t)bs * n * n,
                 hipMemcpyDeviceToDevice, stream);
  {
    dim3 g((n + 255) / 256, 1, bs);
    k_trsm_lo<<<g, 256, 0, stream>>>(ws + oKxx, ws + oS, n, n,
                                     (size_t)n * n, (size_t)n * n);
    k_trsm_up<<<g, 256, 0, stream>>>(ws + oKxx, ws + oS, n, n,
                                     (size_t)n * n, (size_t)n * n);
  }
  k_T_from_S<<<dim3((n + 31) / 32, (n + 31) / 32, bs), b32x8, 0, stream>>>(
      ws + oS, ws + oT, n);
  // X = Kxx^-1 opiK^T  (n x km per batch)
  k_pack_xT<<<dim3((n + 31) / 32, (m + 31) / 32, bs * k), b32x8, 0, stream>>>(
      ws + oOpiK, ws + oX, m, n, k);
  {
    dim3 g((km + 255) / 256, 1, bs);
    k_trsm_lo<<<g, 256, 0, stream>>>(ws + oKxx, ws + oX, n, km,
                                     (size_t)n * n, (size_t)n * km);
    k_trsm_up<<<g, 256, 0, stream>>>(ws + oKxx, ws + oX, n, km,
                                     (size_t)n * n, (size_t)n * km);
  }
  // U_j = sum_i W2[i][j] * X^T_i   (b,k,m,n)
  k_build_U<<<dim3((n + 15) / 16, (m + 15) / 16, bs * k), b16, 0, stream>>>(
      ws + oX, ws + oW2, ws + oU, m, n, k);
  // opimean = opiK @ solvedmean   (km x cin), then Amean
  k_gemm<false, false><<<dim3((cin + 63) / 64, (km + 63) / 64, bs), 256, 0, stream>>>(
      ws + oOpiK, ws + oSM, ws + oOM, km, cin, n, n, cin, cin,
      (long)km * n, (long)n * cin, (long)km * cin);
  {
    int total = bs * m * cout;
    k_amean<<<(total + 255) / 256, 256, 0, stream>>>(ws + oOM, weight, bias,
                                                     outAmean, bs, m, cin, k, cout);
  }
  // R_j = U_j @ T, stored column-block j*n of (m x km) per batch (X reused)
  for (int j = 0; j < k; ++j) {
    k_gemm<false, false><<<dim3((n + 63) / 64, (m + 63) / 64, bs), 256, 0, stream>>>(
        ws + oU + (size_t)j * m * n, ws + oT, Rbuf + (size_t)j * n,
        m, n, n, n, n, km, (long)k * m * n, (long)n * n, (long)m * km);
  }
  // AKA base: closed-form opiKopjfull contraction + jitter*I
  k_copj<<<dim3((n + 15) / 16, (n + 15) / 16, bs), b16, 0, stream>>>(
      xin, ws + oW2, outAKA, n);
  // AKA += sum_j R_j @ opiK_j^T   (sequential accumulating launches; no races)
  for (int j = 0; j < k; ++j) {
    k_gemm<true, true><<<dim3((m + 63) / 64, (m + 63) / 64, bs), 256, 0, stream>>>(
        Rbuf + (size_t)j * n, ws + oOpiK + (size_t)j * m * n, outAKA,
        m, m, n, km, n, m, (long)m * km, (long)k * m * n, (long)m * m);
  }
}